// GATLayer_83227876261958
// MI455X (gfx1250) — compile-verified
//
#include <hip/hip_runtime.h>
#include <hip/hip_bf16.h>
#include <math.h>

#define NEG_SLOPE 0.01f
#define IN_DIM  128
#define OUT_DIM 32

typedef __attribute__((ext_vector_type(2))) float v2f;
typedef __attribute__((ext_vector_type(8))) float v8f;

// ---------------------------------------------------------------------------
// Kernel 0: init scratch + output.  gid over N*32: zero h_out; first N lanes
// also init e_max=-inf, denom=0.
// ---------------------------------------------------------------------------
__global__ void gat_init_kernel(float* __restrict__ h_out,
                                float* __restrict__ e_max,
                                float* __restrict__ denom,
                                int N) {
    int gid = blockIdx.x * blockDim.x + threadIdx.x;
    if (gid < N * OUT_DIM) h_out[gid] = 0.0f;
    if (gid < N) {
        e_max[gid] = -INFINITY;
        denom[gid] = 0.0f;
    }
}

// ---------------------------------------------------------------------------
// Kernel 1: z = h @ fc_w^T via V_WMMA_F32_16X16X4_F32 (native fp32 WMMA).
// One wave32 per 16-row tile of z; two 16x16 f32 accumulators cover OUT=32.
//
// A-frag (16x4 f32, 2 VGPR):  lane L<16 -> row m0+L, K={k,k+1}
//                             lane L>=16 -> row m0+L-16, K={k+2,k+3}
// B-frag (4x16 f32, 2 VGPR):  B[k][n] = fc_w[n][k]; same K half-split by lane.
// C/D (16x16 f32, 8 VGPR):    vgpr g -> row m0 + g + 8*(lane>>4), col lane&15.
// ---------------------------------------------------------------------------
__global__ void gat_gemm_wmma_kernel(const float* __restrict__ h,
                                     const float* __restrict__ fc_w,
                                     float* __restrict__ z,
                                     int N) {
    const int lane   = threadIdx.x & 31;
    const int waveId = blockIdx.x * (blockDim.x >> 5) + (threadIdx.x >> 5);
    const int m0     = waveId * 16;
    if (m0 >= N) return;                       // uniform per-wave: EXEC stays full

    const int n15   = lane & 15;               // row (A) / col (B,D) within tile
    const int khalf = lane >> 4;               // K half-split select

    const float* __restrict__ arow = h + (size_t)(m0 + n15) * IN_DIM + 2 * khalf;
    const float* __restrict__ b0p  = fc_w + (size_t)n15        * IN_DIM + 2 * khalf;
    const float* __restrict__ b1p  = fc_w + (size_t)(16 + n15) * IN_DIM + 2 * khalf;

    v8f acc0 = {};   // cols 0..15
    v8f acc1 = {};   // cols 16..31

    #pragma unroll
    for (int k = 0; k < IN_DIM; k += 4) {
        v2f a  = *(const v2f*)(arow + k);
        v2f b0 = *(const v2f*)(b0p + k);
        v2f b1 = *(const v2f*)(b1p + k);
        acc0 = __builtin_amdgcn_wmma_f32_16x16x4_f32(
                   false, a, false, b0, (short)0, acc0, false, false);
        acc1 = __builtin_amdgcn_wmma_f32_16x16x4_f32(
                   false, a, false, b1, (short)0, acc1, false, false);
    }

    // Scatter accumulators per documented C/D layout.
    #pragma unroll
    for (int g = 0; g < 8; ++g) {
        int row = m0 + g + 8 * khalf;
        z[(size_t)row * OUT_DIM + n15]      = acc0[g];
        z[(size_t)row * OUT_DIM + 16 + n15] = acc1[g];
    }
}

// ---------------------------------------------------------------------------
// Kernel 2: s_l[i] = z[i,:] . a_l ; s_r[i] = z[i,:] . a_r   (one thread/node)
// ---------------------------------------------------------------------------
__global__ void gat_sls_kernel(const float* __restrict__ z,
                               const float* __restrict__ attn_w,  // 64 floats
                               float* __restrict__ s_l,
                               float* __restrict__ s_r,
                               int N) {
    int i = blockIdx.x * blockDim.x + threadIdx.x;
    if (i >= N) return;
    const float4* zr = (const float4*)(z + (size_t)i * OUT_DIM);
    const float4* al = (const float4*)(attn_w);
    const float4* ar = (const float4*)(attn_w + OUT_DIM);
    float accl = 0.0f, accr = 0.0f;
    #pragma unroll
    for (int q = 0; q < OUT_DIM / 4; ++q) {
        float4 zv = zr[q];
        float4 lv = al[q];
        float4 rv = ar[q];
        accl += zv.x * lv.x + zv.y * lv.y + zv.z * lv.z + zv.w * lv.w;
        accr += zv.x * rv.x + zv.y * rv.y + zv.z * rv.z + zv.w * rv.w;
    }
    s_l[i] = accl;
    s_r[i] = accr;
}

// ---------------------------------------------------------------------------
// Kernel 3: per edge, e = leaky_relu(s_l[src]+s_r[dst]); atomic segment max.
// ---------------------------------------------------------------------------
__global__ void gat_edge_max_kernel(const int* __restrict__ src,
                                    const int* __restrict__ dst,
                                    const float* __restrict__ s_l,
                                    const float* __restrict__ s_r,
                                    float* __restrict__ e_buf,
                                    float* __restrict__ e_max,
                                    int E) {
    int i = blockIdx.x * blockDim.x + threadIdx.x;
    if (i >= E) return;
    int s = src[i], t = dst[i];
    float ev = s_l[s] + s_r[t];
    ev = ev > 0.0f ? ev : NEG_SLOPE * ev;
    e_buf[i] = ev;
    atomicMax(&e_max[t], ev);                  // GLOBAL_ATOMIC_MAX_NUM_F32
}

// ---------------------------------------------------------------------------
// Kernel 4: per edge, exp_e = exp(e - e_max[dst]) (in place); segment sum.
// ---------------------------------------------------------------------------
__global__ void gat_edge_exp_kernel(const int* __restrict__ dst,
                                    float* __restrict__ e_buf,
                                    const float* __restrict__ e_max,
                                    float* __restrict__ denom,
                                    int E) {
    int i = blockIdx.x * blockDim.x + threadIdx.x;
    if (i >= E) return;
    int t = dst[i];
    float ex = __expf(e_buf[i] - e_max[t]);
    e_buf[i] = ex;
    atomicAdd(&denom[t], ex);
}

// ---------------------------------------------------------------------------
// Kernel 5: wave-per-edge scatter. Lane d of the wave handles feature dim d,
// so z[src,:] gathers and h_out[dst,:] atomics are contiguous 128B per wave.
// ---------------------------------------------------------------------------
__global__ void gat_scatter_kernel(const int* __restrict__ src,
                                   const int* __restrict__ dst,
                                   const float* __restrict__ z,
                                   const float* __restrict__ e_buf,
                                   const float* __restrict__ denom,
                                   float* __restrict__ h_out,
                                   int E) {
    int gid  = blockIdx.x * blockDim.x + threadIdx.x;
    int edge = gid >> 5;                 // one wave32 per edge
    int d    = gid & 31;                 // lane -> feature dim
    if (edge >= E) return;
    int s = src[edge], t = dst[edge];
    float alpha = e_buf[edge] / denom[t];         // broadcast loads, L2-resident
    float val   = alpha * z[(size_t)s * OUT_DIM + d];
    atomicAdd(&h_out[(size_t)t * OUT_DIM + d], val);
}

// ---------------------------------------------------------------------------
extern "C" void kernel_launch(void* const* d_in, const int* in_sizes, int n_in,
                              void* d_out, int out_size, void* d_ws, size_t ws_size,
                              hipStream_t stream) {
    const float* h      = (const float*)d_in[0];
    const int*   src    = (const int*)  d_in[1];
    const int*   dst    = (const int*)  d_in[2];
    const float* fc_w   = (const float*)d_in[3];
    const float* attn_w = (const float*)d_in[4];

    const int N = in_sizes[0] / IN_DIM;    // 100000
    const int E = in_sizes[1];             // 1600000

    float* h_out = (float*)d_out;

    // Carve workspace (256B aligned chunks).
    auto align256 = [](size_t x) { return (x + 255) & ~(size_t)255; };
    char* w = (char*)d_ws;
    float* z     = (float*)w; w += align256((size_t)N * OUT_DIM * sizeof(float));
    float* s_l   = (float*)w; w += align256((size_t)N * sizeof(float));
    float* s_r   = (float*)w; w += align256((size_t)N * sizeof(float));
    float* e_max = (float*)w; w += align256((size_t)N * sizeof(float));
    float* denom = (float*)w; w += align256((size_t)N * sizeof(float));
    float* e_buf = (float*)w; w += align256((size_t)E * sizeof(float));
    (void)ws_size;

    const int B = 256;

    // 0) init
    {
        int total = N * OUT_DIM;
        gat_init_kernel<<<(total + B - 1) / B, B, 0, stream>>>(h_out, e_max, denom, N);
    }
    // 1) z = h @ fc_w^T  (WMMA, one wave per 16-row tile, 8 waves per block)
    {
        int tiles = (N + 15) / 16;
        int wavesPerBlock = B / 32;
        int grid = (tiles + wavesPerBlock - 1) / wavesPerBlock;
        gat_gemm_wmma_kernel<<<grid, B, 0, stream>>>(h, fc_w, z, N);
    }
    // 2) s_l / s_r
    gat_sls_kernel<<<(N + B - 1) / B, B, 0, stream>>>(z, attn_w, s_l, s_r, N);
    // 3) edge leaky-relu + segment max
    gat_edge_max_kernel<<<(E + B - 1) / B, B, 0, stream>>>(src, dst, s_l, s_r,
                                                           e_buf, e_max, E);
    // 4) exp + segment sum
    gat_edge_exp_kernel<<<(E + B - 1) / B, B, 0, stream>>>(dst, e_buf, e_max,
                                                           denom, E);
    // 5) normalize + scatter-add (wave per edge)
    {
        long long total = (long long)E * 32;
        int grid = (int)((total + B - 1) / B);
        gat_scatter_kernel<<<grid, B, 0, stream>>>(src, dst, z, e_buf, denom,
                                                   h_out, E);
    }
}